// LGL_Block_15126874817050
// MI455X (gfx1250) — compile-verified
//
#include <hip/hip_runtime.h>
#include <math.h>

typedef __attribute__((ext_vector_type(2))) float v2f;
typedef __attribute__((ext_vector_type(4))) float v4f;
typedef __attribute__((ext_vector_type(8))) float v8f;

#define BATCH 2048
#define DIMSZ 512
#define NCLS  62

// Inverse map: output class c (0..61) -> (group i, channel j within x_i)
__device__ __constant__ int d_c2i[62] = {
  0,0,0,0,0, 1, 0,0,0,0,0,0,0, 1,1,1, 2,2,2,2,2, 1,1,1,1, 2,2,2,2,2,
  1,1,1,1, 2,2,2,2,2, 1,1,1,1, 2,2,2,2,2, 1,1, 3,3,3,3,3,3,3,3,3,3,3,3 };
__device__ __constant__ int d_c2j[62] = {
  1,0,2,3,4, 0, 5,6,7,8,9,10,11, 1,2,3, 0,1,2,3,4, 4,5,6,7, 5,6,7,8,9,
  8,9,10,11, 10,11,12,13,14, 12,13,14,15, 15,16,17,18,19, 16,17,
  0,1,2,3,4,5,6,7,8,9,10,11 };

__device__ __forceinline__ float wred_max(float v) {
  #pragma unroll
  for (int o = 16; o > 0; o >>= 1) v = fmaxf(v, __shfl_xor(v, o, 32));
  return v;
}
__device__ __forceinline__ float wred_sum(float v) {
  #pragma unroll
  for (int o = 16; o > 0; o >>= 1) v += __shfl_xor(v, o, 32);
  return v;
}

// ---------------- Kernel A: confidence + 4-way softmax --------------------
// one wave32 per batch row; 8 waves per block
__global__ void conf_kernel(const float* __restrict__ l1, const float* __restrict__ l2,
                            const float* __restrict__ l3, const float* __restrict__ l4,
                            float* __restrict__ out, float* __restrict__ nc_ws) {
  const int lane = threadIdx.x & 31;
  const int wave = threadIdx.x >> 5;
  const int b = blockIdx.x * 8 + wave;

  float conf[4];
  #pragma unroll
  for (int li = 0; li < 4; ++li) {
    const float* lg = (li == 0 ? l1 : li == 1 ? l2 : li == 2 ? l3 : l4) + (size_t)b * NCLS;
    const bool hv = (lane < NCLS - 32);           // lane+32 valid
    float v0 = lg[lane];
    float v1 = hv ? lg[lane + 32] : -3.402823e38f;
    float mx = wred_max(fmaxf(v0, v1));
    float e0 = expf(v0 - mx);
    float e1 = hv ? expf(v1 - mx) : 0.0f;
    float s  = wred_sum(e0 + e1);
    float inv = 1.0f / s;                          // == max_p (max logit -> exp(0)/s)
    float p0 = e0 * inv, p1 = e1 * inv;
    float ec = p0 * logf(p0 + 1e-8f) + (hv ? p1 * logf(p1 + 1e-8f) : 0.0f);
    float ent = -wred_sum(ec);
    float norm_ent = ent * (1.0f / logf((float)NCLS));
    conf[li] = inv * (1.0f - norm_ent);
  }
  // softmax over the 4 confidences (all lanes hold identical values)
  float m4 = fmaxf(fmaxf(conf[0], conf[1]), fmaxf(conf[2], conf[3]));
  float e[4], s4 = 0.0f;
  #pragma unroll
  for (int i = 0; i < 4; ++i) { e[i] = expf(conf[i] - m4); s4 += e[i]; }
  float inv4 = 1.0f / s4;
  if (lane == 0) {
    #pragma unroll
    for (int i = 0; i < 4; ++i) {
      float v = e[i] * inv4;
      out[(size_t)i * BATCH + b] = v;   // outputs 0..3: norm_conf columns
      nc_ws[b * 4 + i] = v;
    }
  }
}

// ---------------- Kernel B: GEMV via V_WMMA_F32_16X16X4_F32 ---------------
// 16 batch rows per block; 8 waves each own a K/8 slice (scalar bounds);
// 4x-unrolled load/WMMA pipeline + prefetch; LDS cross-wave reduce.
__global__ void dot_u_kernel(const float* __restrict__ x, const float* __restrict__ W,
                             const float* __restrict__ bias, const float* __restrict__ nc,
                             float* __restrict__ uf, int K, int grp) {
  const int lane = threadIdx.x & 31;
  const int wave = __builtin_amdgcn_readfirstlane(threadIdx.x >> 5);  // SGPR -> scalar loop
  const int row  = (blockIdx.x << 4) + (lane & 15);
  const float* xrow = x + (size_t)row * K;
  const int kofs = (lane >> 4) << 1;          // lanes 0-15 -> K 0,1 ; lanes 16-31 -> K 2,3
  const int Kw = K >> 3;                      // per-wave K slice (multiple of 16 for all groups)
  const int k0 = wave * Kw, k1 = k0 + Kw;

  v8f acc = {0.f, 0.f, 0.f, 0.f, 0.f, 0.f, 0.f, 0.f};
  #pragma unroll 1
  for (int k = k0; k < k1; k += 16) {
    // prefetch ~2KB ahead in each row's stream (global_prefetch_b8)
    if (k + 512 < k1) __builtin_prefetch(xrow + k + 512, 0, 1);
    v2f a0 = *(const v2f*)(xrow + k      + kofs);
    v2f b0 = *(const v2f*)(W    + k      + kofs);
    v2f a1 = *(const v2f*)(xrow + k + 4  + kofs);
    v2f b1 = *(const v2f*)(W    + k + 4  + kofs);
    v2f a2 = *(const v2f*)(xrow + k + 8  + kofs);
    v2f b2 = *(const v2f*)(W    + k + 8  + kofs);
    v2f a3 = *(const v2f*)(xrow + k + 12 + kofs);
    v2f b3 = *(const v2f*)(W    + k + 12 + kofs);
    acc = __builtin_amdgcn_wmma_f32_16x16x4_f32(false, a0, false, b0, (short)0, acc, false, false);
    acc = __builtin_amdgcn_wmma_f32_16x16x4_f32(false, a1, false, b1, (short)0, acc, false, false);
    acc = __builtin_amdgcn_wmma_f32_16x16x4_f32(false, a2, false, b2, (short)0, acc, false, false);
    acc = __builtin_amdgcn_wmma_f32_16x16x4_f32(false, a3, false, b3, (short)0, acc, false, false);
  }

  // D[m,n] identical across n; lanes 0 (rows 0..7) and 16 (rows 8..15) publish.
  __shared__ float red[16 * 8];
  if ((lane & 15) == 0) {
    const int rbase = (lane >> 4) << 3;
    #pragma unroll
    for (int r = 0; r < 8; ++r) red[((rbase + r) << 3) + wave] = acc[r];
  }
  __syncthreads();
  if (threadIdx.x < 16) {
    float s = 0.0f;
    #pragma unroll
    for (int w = 0; w < 8; ++w) s += red[(threadIdx.x << 3) + w];
    const int b = (blockIdx.x << 4) + threadIdx.x;
    float u = 1.0f / (1.0f + expf(-(s + bias[0])));
    uf[b * 4 + grp] = u * nc[b * 4 + grp];    // fold norm_conf in now
  }
}

// ---------------- Kernel C: trans = (x + g_region) * u_final --------------
// 2 rows of 512 floats per 256-thread block; 16B vector streams.
// g is single-use -> NT load; trans is write-once -> NT store; x stays RT so
// L2 (192MB) can serve it from the preceding dot_u pass.
__global__ void trans_kernel(const float* __restrict__ x1, const float* __restrict__ x2,
                             const float* __restrict__ x3, const float* __restrict__ x4,
                             const float* __restrict__ g,  const float* __restrict__ uf,
                             float* __restrict__ outT) {
  const int rc = blockIdx.x * 2 + (threadIdx.x >> 7);   // row index in [0, B*62)
  const int t  = threadIdx.x & 127;                     // 16B-vector index 0..127
  const int b  = rc / NCLS;
  const int c  = rc - b * NCLS;
  const int i  = d_c2i[c];
  const int j  = d_c2j[c];
  const float* xp = (i == 0 ? x1 : i == 1 ? x2 : i == 2 ? x3 : x4);
  const int    ch = (i == 0 ? 12 : i == 1 ? 18 : i == 2 ? 20 : 12);

  const v4f xv = ((const v4f*)(xp + ((size_t)b * ch + j) * DIMSZ))[t];
  const v4f gv = __builtin_nontemporal_load((const v4f*)(g + (size_t)rc * DIMSZ) + t);
  const float u = uf[b * 4 + i];
  v4f o = (xv + gv) * u;
  __builtin_nontemporal_store(o, (v4f*)(outT + (size_t)rc * DIMSZ) + t);
}

extern "C" void kernel_launch(void* const* d_in, const int* in_sizes, int n_in,
                              void* d_out, int out_size, void* d_ws, size_t ws_size,
                              hipStream_t stream) {
  (void)in_sizes; (void)n_in; (void)out_size; (void)ws_size;
  const float* x[4]  = {(const float*)d_in[0], (const float*)d_in[1],
                        (const float*)d_in[2], (const float*)d_in[3]};
  const float* g     = (const float*)d_in[4];
  const float* lg[4] = {(const float*)d_in[5], (const float*)d_in[6],
                        (const float*)d_in[7], (const float*)d_in[8]};
  const float* W[4]  = {(const float*)d_in[9],  (const float*)d_in[11],
                        (const float*)d_in[13], (const float*)d_in[15]};
  const float* bs[4] = {(const float*)d_in[10], (const float*)d_in[12],
                        (const float*)d_in[14], (const float*)d_in[16]};
  float* out   = (float*)d_out;
  float* nc_ws = (float*)d_ws;            // norm_conf, B*4 floats
  float* uf_ws = nc_ws + 4 * BATCH;       // u_final,  B*4 floats
  const int Ks[4] = {12 * DIMSZ, 18 * DIMSZ, 20 * DIMSZ, 12 * DIMSZ};

  conf_kernel<<<BATCH / 8, 256, 0, stream>>>(lg[0], lg[1], lg[2], lg[3], out, nc_ws);
  for (int i = 0; i < 4; ++i)
    dot_u_kernel<<<BATCH / 16, 256, 0, stream>>>(x[i], W[i], bs[i], nc_ws, uf_ws, Ks[i], i);
  trans_kernel<<<(BATCH * NCLS) / 2, 256, 0, stream>>>(
      x[0], x[1], x[2], x[3], g, uf_ws, out + 4 * BATCH);
}